// RPNPostProcessor_26774826123599
// MI455X (gfx1250) — compile-verified
//
#include <hip/hip_runtime.h>
#include <cstdint>
#include <cstddef>

// ---------------- problem constants ----------------
#define NN    8
#define AA    12
#define HH    200
#define WW    304
#define HWc   (HH*WW)        // 60800
#define AHWc  (AA*HWc)       // 729600
#define KK    6000
#define POSTN 1000
#define IMGW  1216.0f
#define IMGH  800.0f
#define NEGV  (-1e30f)
#define BCLIP 4.135166556742356f   // log(1000/16)
#define STG   42000                // floats per image in staging (7 arrays x 6000)
#define NMS_T 1024

// workspace byte offsets
#define OFF_H1   ((size_t)0)
#define OFF_H2   (OFF_H1 + (size_t)NN*2048*4)   // 65536
#define OFF_H3   (OFF_H2 + (size_t)NN*2048*4)   // 131072
#define OFF_META (OFF_H3 + (size_t)NN*1024*4)   // 163840  (64 u32)
#define OFF_EQM  (OFF_META + 256)               // 164096
#define OFF_STG  (OFF_EQM + (size_t)NN*1024*4)  // 196864 (16B aligned)
#define ZERO_WORDS (OFF_STG/4)

// meta u32 slots: b1:0 rem1:8 b2:16 rem2:24 T:32 needeq:40 cntgt:48 cnteq:56

// CDNA5 async global->LDS path (probe via __has_builtin; fallback = plain copy)
#if defined(__HIP_DEVICE_COMPILE__) && __has_builtin(__builtin_amdgcn_global_load_async_to_lds_b128) && __has_builtin(__builtin_amdgcn_s_wait_asynccnt)
#define USE_ASYNC_LDS 1
#else
#define USE_ASYNC_LDS 0
#endif

#if USE_ASYNC_LDS
typedef int v4i_t __attribute__((vector_size(16)));                 // int __vector(4)
typedef __attribute__((address_space(1))) v4i_t gv4i_t;             // global v4i
typedef __attribute__((address_space(3))) v4i_t lv4i_t;             // LDS v4i
#endif

__device__ __forceinline__ unsigned fkey(float f) {
    unsigned u = __float_as_uint(f);
    return (u & 0x80000000u) ? ~u : (u | 0x80000000u); // monotonic: larger float -> larger key
}

__global__ void k_zero(unsigned* p, int n) {
    int i = blockIdx.x * blockDim.x + threadIdx.x;
    if (i < n) p[i] = 0u;
}

// ---------------- pass 1: 2048-bin histogram of top 11 key bits ----------------
#define H1_ELEMS 16
__global__ void k_hist1(const float* __restrict__ obj, unsigned* __restrict__ hist1) {
    __shared__ unsigned sh[2048];
    const int n = blockIdx.y;
    for (int i = threadIdx.x; i < 2048; i += blockDim.x) sh[i] = 0u;
    __syncthreads();
    const float* p = obj + (size_t)n * AHWc;
    const int base = blockIdx.x * (blockDim.x * H1_ELEMS);
    for (int e = 0; e < H1_ELEMS; ++e) {
        int m = base + e * blockDim.x + threadIdx.x;
        if (m < AHWc) {
            int mp = m + blockDim.x * H1_ELEMS;               // stream-ahead prefetch
            if (mp < AHWc) __builtin_prefetch(p + mp, 0, 1);  // -> global_prefetch_b8
            unsigned k = fkey(p[m]);
            atomicAdd(&sh[k >> 21], 1u);
        }
    }
    __syncthreads();
    for (int i = threadIdx.x; i < 2048; i += blockDim.x)
        if (sh[i]) atomicAdd(&hist1[n * 2048 + i], sh[i]);
}

__global__ void k_sel1(const unsigned* __restrict__ hist, unsigned* __restrict__ meta) {
    int n = blockIdx.x; if (threadIdx.x) return;
    const unsigned* h = hist + n * 2048;
    unsigned cum = 0; int b = 2047;
    for (; b >= 0; --b) { unsigned c = h[b]; if (cum + c >= KK) break; cum += c; }
    meta[0 + n] = (unsigned)b; meta[8 + n] = KK - cum;
}

__global__ void k_hist2(const float* __restrict__ obj, const unsigned* __restrict__ meta,
                        unsigned* __restrict__ hist2) {
    int n = blockIdx.y; int m = blockIdx.x * 256 + threadIdx.x; if (m >= AHWc) return;
    unsigned k = fkey(obj[(size_t)n * AHWc + m]);
    if ((k >> 21) == meta[0 + n]) atomicAdd(&hist2[n * 2048 + ((k >> 10) & 2047u)], 1u);
}

__global__ void k_sel2(const unsigned* __restrict__ hist, unsigned* __restrict__ meta) {
    int n = blockIdx.x; if (threadIdx.x) return;
    unsigned need = meta[8 + n];
    const unsigned* h = hist + n * 2048;
    unsigned cum = 0; int b = 2047;
    for (; b >= 0; --b) { unsigned c = h[b]; if (cum + c >= need) break; cum += c; }
    meta[16 + n] = (unsigned)b; meta[24 + n] = need - cum;
}

__global__ void k_hist3(const float* __restrict__ obj, const unsigned* __restrict__ meta,
                        unsigned* __restrict__ hist3) {
    int n = blockIdx.y; int m = blockIdx.x * 256 + threadIdx.x; if (m >= AHWc) return;
    unsigned k = fkey(obj[(size_t)n * AHWc + m]);
    unsigned pre = (meta[0 + n] << 11) | meta[16 + n];
    if ((k >> 10) == pre) atomicAdd(&hist3[n * 1024 + (k & 1023u)], 1u);
}

__global__ void k_sel3(const unsigned* __restrict__ hist, unsigned* __restrict__ meta) {
    int n = blockIdx.x; if (threadIdx.x) return;
    unsigned need = meta[24 + n];
    const unsigned* h = hist + n * 1024;
    unsigned cum = 0; int b = 1023;
    for (; b >= 0; --b) { unsigned c = h[b]; if (cum + c >= need) break; cum += c; }
    meta[32 + n] = (meta[0 + n] << 21) | (meta[16 + n] << 10) | (unsigned)b;  // exact threshold key
    meta[40 + n] = need - cum;                                               // #equal-key to take
}

// decode + clip + keep-mask + sigmoid, write SoA staging slot
__device__ __forceinline__ void decode_write(int n, int m, int slot,
                                             const float* __restrict__ anchors,
                                             const float* __restrict__ obj,
                                             const float* __restrict__ reg,
                                             float* __restrict__ stage) {
    int a = m / HWc; int r = m - a * HWc;
    int li = r * AA + a;  // logical flatten index (matches permute_and_flatten)
    const float* an = anchors + ((size_t)n * AHWc + (size_t)li) * 4;
    float ax1 = an[0], ay1 = an[1], ax2 = an[2], ay2 = an[3];
    float aw = ax2 - ax1 + 1.0f, ah = ay2 - ay1 + 1.0f;
    float acx = ax1 + 0.5f * aw, acy = ay1 + 0.5f * ah;
    size_t rb = ((size_t)(n * AA + a) * 4) * HWc + (size_t)r;
    float dx = reg[rb], dy = reg[rb + HWc];
    float dw = fminf(reg[rb + 2 * (size_t)HWc], BCLIP);
    float dh = fminf(reg[rb + 3 * (size_t)HWc], BCLIP);
    float pcx = dx * aw + acx, pcy = dy * ah + acy;
    float pw = expf(dw) * aw, ph = expf(dh) * ah;
    float x1 = pcx - 0.5f * pw, y1 = pcy - 0.5f * ph;
    float x2 = pcx + 0.5f * pw - 1.0f, y2 = pcy + 0.5f * ph - 1.0f;
    x1 = fminf(fmaxf(x1, 0.0f), IMGW - 1.0f);
    y1 = fminf(fmaxf(y1, 0.0f), IMGH - 1.0f);
    x2 = fminf(fmaxf(x2, 0.0f), IMGW - 1.0f);
    y2 = fminf(fmaxf(y2, 0.0f), IMGH - 1.0f);
    bool keep = ((x2 - x1 + 1.0f) >= 0.0f) && ((y2 - y1 + 1.0f) >= 0.0f);
    float logit = obj[(size_t)n * AHWc + m];
    float sc = 1.0f / (1.0f + expf(-logit));
    if (!keep) sc = NEGV;
    float area = (x2 - x1 + 1.0f) * (y2 - y1 + 1.0f);
    float* st = stage + (size_t)n * STG;
    st[slot] = x1; st[6000 + slot] = y1; st[12000 + slot] = x2; st[18000 + slot] = y2;
    st[24000 + slot] = area; st[30000 + slot] = sc;
    st[36000 + slot] = __uint_as_float((unsigned)li);
}

__global__ void k_compact(const float* __restrict__ anchors, const float* __restrict__ obj,
                          const float* __restrict__ reg, unsigned* __restrict__ meta,
                          unsigned* __restrict__ eqm, float* __restrict__ stage) {
    int n = blockIdx.y; int m = blockIdx.x * 256 + threadIdx.x; if (m >= AHWc) return;
    unsigned k = fkey(obj[(size_t)n * AHWc + m]);
    unsigned t = meta[32 + n];
    if (k > t) {
        unsigned slot = atomicAdd(&meta[48 + n], 1u);   // in [0, KK-need_eq)
        decode_write(n, m, (int)slot, anchors, obj, reg, stage);
    } else if (k == t) {
        unsigned e = atomicAdd(&meta[56 + n], 1u);
        if (e < 1024u) eqm[n * 1024 + e] = (unsigned)m;
    }
}

// deterministic resolution of equal-key elements: take smallest logical indices (top_k tie rule)
__global__ void k_eqresolve(const float* __restrict__ anchors, const float* __restrict__ obj,
                            const float* __restrict__ reg, const unsigned* __restrict__ meta,
                            const unsigned* __restrict__ eqm, float* __restrict__ stage) {
    int n = blockIdx.x; if (threadIdx.x != 0) return;
    unsigned need = meta[40 + n];
    unsigned c = meta[56 + n]; if (c > 1024u) c = 1024u;
    unsigned base = KK - need;
    unsigned take = need < c ? need : c;
    unsigned lm[1024], lli[1024];
    for (unsigned i = 0; i < c; ++i) {
        unsigned m = eqm[n * 1024 + i]; lm[i] = m;
        unsigned a = m / HWc, r = m - a * HWc; lli[i] = r * AA + a;
    }
    for (unsigned j = 0; j < take; ++j) {
        unsigned bi = 0, bv = 0xFFFFFFFFu;
        for (unsigned i = 0; i < c; ++i) if (lli[i] < bv) { bv = lli[i]; bi = i; }
        decode_write(n, (int)lm[bi], (int)(base + j), anchors, obj, reg, stage);
        lli[bi] = 0xFFFFFFFFu;
    }
    float* st = stage + (size_t)n * STG;   // (statistically unreachable) fill
    for (unsigned j = take; j < need; ++j) {
        int s = (int)(base + j);
        st[s] = 0; st[6000 + s] = 0; st[12000 + s] = 0; st[18000 + s] = 0;
        st[24000 + s] = 1.0f; st[30000 + s] = NEGV;
        st[36000 + s] = __uint_as_float(0x7FFFFFFFu);
    }
}

// ---------------- NMS: 1 workgroup per image, full working set in LDS (168KB) ----------------
__global__ __launch_bounds__(NMS_T) void k_nms(const float* __restrict__ stage, float* __restrict__ out) {
    extern __shared__ float lds[];
    const int n = blockIdx.x, tid = threadIdx.x;
    const float* src = stage + (size_t)n * STG;
#if USE_ASYNC_LDS
    for (int c4 = tid; c4 < STG / 4; c4 += NMS_T) {
        gv4i_t* g = (gv4i_t*)(uintptr_t)(const void*)(src + c4 * 4);
        lv4i_t* l = (lv4i_t*)(unsigned)(uintptr_t)(const void*)(lds + c4 * 4);
        __builtin_amdgcn_global_load_async_to_lds_b128(g, l, 0, 0);
    }
    __builtin_amdgcn_s_wait_asynccnt(0);
#else
    {
        const float4* g4 = (const float4*)src; float4* l4 = (float4*)lds;
        for (int c4 = tid; c4 < STG / 4; c4 += NMS_T) l4[c4] = g4[c4];
    }
#endif
    __syncthreads();
    float* X1 = lds;           float* Y1 = lds + 6000;  float* X2 = lds + 12000;
    float* Y2 = lds + 18000;   float* AR = lds + 24000; float* SC = lds + 30000;
    float* IDv = lds + 36000;
    __shared__ float rs[32]; __shared__ unsigned ri[32]; __shared__ int rj[32];
    __shared__ float s_best; __shared__ int s_slot;
    const int lane = tid & 31, wv = tid >> 5;   // wave32
    int it = 0;
    for (; it < POSTN; ++it) {
        float bs = -3.4e38f; unsigned bi = 0xFFFFFFFFu; int bj = 0;
        for (int e = tid; e < KK; e += NMS_T) {
            float s = SC[e]; unsigned id = __float_as_uint(IDv[e]);
            if (s > bs || (s == bs && id < bi)) { bs = s; bi = id; bj = e; }
        }
        for (int o = 16; o > 0; o >>= 1) {      // wave32 shuffle reduction
            float os = __shfl_xor(bs, o, 32);
            unsigned oi = __shfl_xor(bi, o, 32);
            int oj = __shfl_xor(bj, o, 32);
            if (os > bs || (os == bs && oi < bi)) { bs = os; bi = oi; bj = oj; }
        }
        if (lane == 0) { rs[wv] = bs; ri[wv] = bi; rj[wv] = bj; }
        __syncthreads();
        if (wv == 0) {
            float s = rs[lane]; unsigned i2 = ri[lane]; int j2 = rj[lane];
            for (int o = 16; o > 0; o >>= 1) {
                float os = __shfl_xor(s, o, 32);
                unsigned oi = __shfl_xor(i2, o, 32);
                int oj = __shfl_xor(j2, o, 32);
                if (os > s || (os == s && oi < i2)) { s = os; i2 = oi; j2 = oj; }
            }
            if (lane == 0) { s_best = s; s_slot = j2; }
        }
        __syncthreads();
        float best = s_best; int j = s_slot;
        if (best <= -5e29f) break;   // all remaining invalid -> zero-filled below
        float jx1 = X1[j], jy1 = Y1[j], jx2 = X2[j], jy2 = Y2[j], jar = AR[j];
        if (tid == 0) {
            float* o = out + ((size_t)n * POSTN + it) * 5;
            o[0] = jx1; o[1] = jy1; o[2] = jx2; o[3] = jy2; o[4] = best;
        }
        for (int e = tid; e < KK; e += NMS_T) {
            float ltx = fmaxf(jx1, X1[e]), lty = fmaxf(jy1, Y1[e]);
            float rbx = fminf(jx2, X2[e]), rby = fminf(jy2, Y2[e]);
            float w = fmaxf(rbx - ltx + 1.0f, 0.0f), h = fmaxf(rby - lty + 1.0f, 0.0f);
            float inter = w * h;
            float iou = inter / (jar + AR[e] - inter);
            if (iou > 0.7f) SC[e] = NEGV;   // kills j too (iou==1), as in reference
        }
        __syncthreads();
    }
    for (int r = it * 5 + tid; r < POSTN * 5; r += NMS_T)
        out[(size_t)n * POSTN * 5 + r] = 0.0f;
}

extern "C" void kernel_launch(void* const* d_in, const int* in_sizes, int n_in,
                              void* d_out, int out_size, void* d_ws, size_t ws_size,
                              hipStream_t stream) {
    const float* anchors = (const float*)d_in[0];   // [N, AHW, 4]
    const float* obj     = (const float*)d_in[1];   // [N, A, H, W]
    const float* reg     = (const float*)d_in[2];   // [N, A*4, H, W]
    float* out = (float*)d_out;                     // [N, 1000, 5]
    char* ws = (char*)d_ws;
    unsigned* hist1 = (unsigned*)(ws + OFF_H1);
    unsigned* hist2 = (unsigned*)(ws + OFF_H2);
    unsigned* hist3 = (unsigned*)(ws + OFF_H3);
    unsigned* meta  = (unsigned*)(ws + OFF_META);
    unsigned* eqm   = (unsigned*)(ws + OFF_EQM);
    float*    stage = (float*)(ws + OFF_STG);

    k_zero<<<((int)ZERO_WORDS + 255) / 256, 256, 0, stream>>>((unsigned*)ws, (int)ZERO_WORDS);

    dim3 g1((AHWc + 256 * H1_ELEMS - 1) / (256 * H1_ELEMS), NN);
    k_hist1<<<g1, 256, 0, stream>>>(obj, hist1);
    k_sel1<<<NN, 1, 0, stream>>>(hist1, meta);

    dim3 g2((AHWc + 255) / 256, NN);
    k_hist2<<<g2, 256, 0, stream>>>(obj, meta, hist2);
    k_sel2<<<NN, 1, 0, stream>>>(hist2, meta);
    k_hist3<<<g2, 256, 0, stream>>>(obj, meta, hist3);
    k_sel3<<<NN, 1, 0, stream>>>(hist3, meta);

    k_compact<<<g2, 256, 0, stream>>>(anchors, obj, reg, meta, eqm, stage);
    k_eqresolve<<<NN, 32, 0, stream>>>(anchors, obj, reg, meta, eqm, stage);

    k_nms<<<NN, NMS_T, STG * sizeof(float), stream>>>(stage, out);
}